// edge_processer_28647431864537
// MI455X (gfx1250) — compile-verified
//
#include <hip/hip_runtime.h>
#include <hip/hip_bf16.h>

typedef __attribute__((ext_vector_type(2))) float v2f;
typedef __attribute__((ext_vector_type(8))) float v8f;

#define IN_DIM 128

// Phase 1: per-node partial dot products via V_WMMA_F32_16X16X4_F32.
// One wave (32 lanes) computes a 16-node tile:
//   D[16x16] = A(16x128 slice of x) x B(128x16), where B col0 = W[0:128],
//   B col1 = W[128:256], other cols 0.  K consumed 4 at a time (32 WMMAs).
// f32 WMMA keeps full fp32 precision (matches the fp32 reference).
// Column masking is done with v_cndmask on loaded values -- NO divergent
// branches in the inner loop (WMMA wants EXEC all-1s; branchless keeps the
// loop at 2 loads + 2 selects + 1 wmma).
__global__ void __launch_bounds__(32)
node_scores_wmma_kernel(const float* __restrict__ x,
                        const float* __restrict__ W,
                        float* __restrict__ s0,
                        float* __restrict__ s1,
                        int n_nodes) {
    const int lane = threadIdx.x;        // 0..31
    const int tile = blockIdx.x;         // 16 nodes per tile
    const int half = lane >> 4;          // 0: lanes 0-15, 1: lanes 16-31
    const int n    = lane & 15;          // B/C column index for this lane
    const bool live_col = (n < 2);       // only cols 0/1 of B are nonzero

    int node = tile * 16 + n;            // A-matrix row fed by this lane
    if (node >= n_nodes) node = n_nodes - 1;   // clamp reads; stores guarded

    // Per-lane base pointers; loop offset is the uniform 4*t.
    const float* __restrict__ ap = x + (size_t)node * IN_DIM + 2 * half;
    // Always-in-bounds W address (lanes n>=2 alias column 0, value masked).
    const float* __restrict__ wp = W + (live_col ? n : 0) * IN_DIM + 2 * half;

    v8f c = {};                          // 16x16 f32 accumulator (8 VGPRs)
    #pragma unroll 4
    for (int t = 0; t < IN_DIM / 4; ++t) {
        const int k = 4 * t;
        // A 16x4 f32: VGPR0 = K, VGPR1 = K+1 (lanes 0-15: K=4t, 16-31: 4t+2)
        v2f a = *(const v2f*)(ap + k);
        // B 4x16 f32: row K striped across lanes; mask dead columns to 0
        // on the VALUE (v_cndmask), never on the load (no EXEC churn).
        v2f w = *(const v2f*)(wp + k);
        v2f b;
        b.x = live_col ? w.x : 0.0f;
        b.y = live_col ? w.y : 0.0f;
        // (neg_a, A, neg_b, B, c_mod, C, reuse_a, reuse_b)
        c = __builtin_amdgcn_wmma_f32_16x16x4_f32(false, a, false, b,
                                                  (short)0, c, false, false);
    }

    // D layout: c[i] @ lanes 0-15 = D[i][lane], @ lanes 16-31 = D[8+i][lane-16].
    // Column 0 -> s0, column 1 -> s1.
    const int rowBase = tile * 16 + half * 8;
    if (n == 0) {
        #pragma unroll
        for (int i = 0; i < 8; ++i)
            if (rowBase + i < n_nodes) s0[rowBase + i] = c[i];
    } else if (n == 1) {
        #pragma unroll
        for (int i = 0; i < 8; ++i)
            if (rowBase + i < n_nodes) s1[rowBase + i] = c[i];
    }
}

// Phase 2: out[e] = s0[src[e]] + s1[dst[e]] + bias.
// edge_index is [2, E] row-major int32: sources at [0..E), targets at [E..2E).
// s0/s1 (80 KB) are L2-resident -> the two random gathers are cheap; the
// edge_index reads and the store are fully coalesced.
__global__ void __launch_bounds__(256)
edge_gather_kernel(const int* __restrict__ edge_index,
                   const float* __restrict__ s0,
                   const float* __restrict__ s1,
                   const float* __restrict__ bias,
                   float* __restrict__ out,
                   int n_edges) {
    const int e = blockIdx.x * blockDim.x + threadIdx.x;
    if (e >= n_edges) return;
    const int src = edge_index[e];
    const int dst = edge_index[n_edges + e];
    out[e] = s0[src] + s1[dst] + bias[0];
}

extern "C" void kernel_launch(void* const* d_in, const int* in_sizes, int n_in,
                              void* d_out, int out_size, void* d_ws, size_t ws_size,
                              hipStream_t stream) {
    const float* x          = (const float*)d_in[0];   // [n_nodes, 128]
    const int*   edge_index = (const int*)  d_in[1];   // [2, n_edges]
    const float* W          = (const float*)d_in[2];   // [1, 256]
    const float* b          = (const float*)d_in[3];   // [1]
    float*       out        = (float*)d_out;           // [n_edges, 1]

    const int n_nodes = in_sizes[0] / IN_DIM;
    const int n_edges = out_size;

    float* s0 = (float*)d_ws;            // n_nodes floats
    float* s1 = s0 + n_nodes;            // n_nodes floats (total 80 KB << ws)

    const int n_tiles = (n_nodes + 15) / 16;
    node_scores_wmma_kernel<<<n_tiles, 32, 0, stream>>>(x, W, s0, s1, n_nodes);

    const int threads = 256;
    const int blocks  = (n_edges + threads - 1) / threads;
    edge_gather_kernel<<<blocks, threads, 0, stream>>>(edge_index, s0, s1, b,
                                                       out, n_edges);
}